// LightGCN_44427141710564
// MI455X (gfx1250) — compile-verified
//
#include <hip/hip_runtime.h>

// ---------------------------------------------------------------------------
// LightGCN on MI455X (gfx1250).
//
// Phases:
//  1) buf0 = concat(user_emb, item_emb)                (d2d memcpy)
//  2) acc_{u,p,n}[b] = ego rows; also write ego outputs (gather kernel)
//  3) 3x: zero(next); next = A * cur (segmented SpMM, sorted-row register
//     accumulation, f32 global atomics only at segment boundaries, scalar
//     edge metadata + global_prefetch of the dependent row gather);
//     acc_{u,p,n} += next[gathered rows]; swap(cur,next)
//  4) scores = diag( (acc_u/4) x (acc_{p,n}/4)^T ) via V_WMMA_F32_16X16X4_F32
//
// Workspace requirement: 2*N*D + 3*B*D floats  (~320 MB for these sizes).
// ---------------------------------------------------------------------------

typedef float v2f __attribute__((ext_vector_type(2)));
typedef float v8f __attribute__((ext_vector_type(8)));

#define D_EMB  256
#define CHUNK  128   // edges per wave in the SpMM
#define PFDIST 8     // edges of prefetch lookahead

// ---------------------------------------------------------------- zero fill
__global__ void zero_f4_kernel(float4* __restrict__ p, long n4) {
  long i      = (long)blockIdx.x * blockDim.x + threadIdx.x;
  long stride = (long)gridDim.x * blockDim.x;
  const float4 z = make_float4(0.f, 0.f, 0.f, 0.f);
  for (; i < n4; i += stride) p[i] = z;
}

// ------------------------------------------------------- segmented SpMM
// One wave per CHUNK contiguous edges. rows[] is sorted within each half of
// the COO list, so consecutive edges mostly share a row: accumulate the
// 256-float row product in VGPRs (lane covers d = lane*4..+3 and 128+lane*4..+3,
// two coalesced b128 loads per edge) and flush with f32 atomics only on
// row change / chunk end. Wave id is forced uniform so the edge metadata
// becomes scalar (SMEM) loads and the row-change branch is a scalar branch.
__global__ void spmm_seg_kernel(const int*   __restrict__ rows,
                                const int*   __restrict__ cols,
                                const float* __restrict__ vals,
                                const float* __restrict__ cur,
                                float*       __restrict__ nxt,
                                int nnz) {
  const int wave = __builtin_amdgcn_readfirstlane(
      (int)(((long)blockIdx.x * blockDim.x + threadIdx.x) >> 5));
  const int lane = threadIdx.x & 31;
  long start = (long)wave * CHUNK;
  if (start >= nnz) return;
  long end = start + CHUNK;
  if (end > nnz) end = nnz;

  float a0x = 0.f, a0y = 0.f, a0z = 0.f, a0w = 0.f;
  float a1x = 0.f, a1y = 0.f, a1z = 0.f, a1w = 0.f;
  int crow = -1;

#define SPMM_FLUSH()                                            \
  do {                                                          \
    float* dst = nxt + (long)crow * D_EMB + lane * 4;           \
    unsafeAtomicAdd(dst + 0,   a0x);                            \
    unsafeAtomicAdd(dst + 1,   a0y);                            \
    unsafeAtomicAdd(dst + 2,   a0z);                            \
    unsafeAtomicAdd(dst + 3,   a0w);                            \
    unsafeAtomicAdd(dst + 128, a1x);                            \
    unsafeAtomicAdd(dst + 129, a1y);                            \
    unsafeAtomicAdd(dst + 130, a1z);                            \
    unsafeAtomicAdd(dst + 131, a1w);                            \
  } while (0)

  for (long e = start; e < end; ++e) {
    // Prefetch the row for edge e+PFDIST: dependent gather into L2; lanes
    // fan out 32B apart so the 32 addresses cover every cacheline of the
    // 1KB row (emits global_prefetch_b8).
    if (e + PFDIST < (long)nnz) {
      const int cpf = cols[e + PFDIST];
      __builtin_prefetch(cur + (long)cpf * D_EMB + lane * 8, 0, 3);
    }

    const int r = rows[e];                 // wave-uniform -> scalar load
    if (r != crow) {                       // scalar compare/branch
      if (crow >= 0) SPMM_FLUSH();
      crow = r;
      a0x = a0y = a0z = a0w = 0.f;
      a1x = a1y = a1z = a1w = 0.f;
    }
    const int   c = cols[e];
    const float v = vals[e];
    const float4* src = (const float4*)(cur + (long)c * D_EMB);
    const float4 x0 = src[lane];           // bytes [lane*16 .. +15]   (coalesced)
    const float4 x1 = src[lane + 32];      // second 512B half of the row
    a0x = fmaf(v, x0.x, a0x);  a0y = fmaf(v, x0.y, a0y);
    a0z = fmaf(v, x0.z, a0z);  a0w = fmaf(v, x0.w, a0w);
    a1x = fmaf(v, x1.x, a1x);  a1y = fmaf(v, x1.y, a1y);
    a1z = fmaf(v, x1.z, a1z);  a1w = fmaf(v, x1.w, a1w);
  }
  if (crow >= 0) SPMM_FLUSH();
#undef SPMM_FLUSH
}

// ------------------------------------------ init accumulators + ego outputs
// wave id -> (table t, batch b). t=0: user, t=1: pos item, t=2: neg item.
__global__ void init_acc_ego_kernel(const float* __restrict__ uemb,
                                    const float* __restrict__ iemb,
                                    const int* __restrict__ userId,
                                    const int* __restrict__ itemId,
                                    const int* __restrict__ negId,
                                    float* __restrict__ accU,
                                    float* __restrict__ accP,
                                    float* __restrict__ accN,
                                    float* __restrict__ ego_out,  // d_out + 2B
                                    int B_) {
  const int wid = __builtin_amdgcn_readfirstlane(
      (int)(((long)blockIdx.x * blockDim.x + threadIdx.x) >> 5));
  const int lane = threadIdx.x & 31;
  if (wid >= 3 * B_) return;
  const int t = wid / B_;
  const int b = wid - t * B_;

  const float* src;
  float* acc;
  float* ego;
  if (t == 0)      { src = uemb + (long)userId[b] * D_EMB; acc = accU; ego = ego_out; }
  else if (t == 1) { src = iemb + (long)itemId[b] * D_EMB; acc = accP; ego = ego_out + (long)B_ * D_EMB; }
  else             { src = iemb + (long)negId[b]  * D_EMB; acc = accN; ego = ego_out + 2l * B_ * D_EMB; }

  const float4* s4 = (const float4*)src;
  float4* a4 = (float4*)(acc + (long)b * D_EMB);
  float4* e4 = (float4*)(ego + (long)b * D_EMB);
  const float4 v0 = s4[lane], v1 = s4[lane + 32];
  a4[lane] = v0;  a4[lane + 32] = v1;   // layer-0 term of the accumulator
  e4[lane] = v0;  e4[lane + 32] = v1;   // ego output
}

// -------------------------------------------------- per-layer gather + add
__global__ void gather_add_kernel(const float* __restrict__ layer,   // N x D
                                  const int* __restrict__ userId,
                                  const int* __restrict__ itemId,
                                  const int* __restrict__ negId,
                                  float* __restrict__ accU,
                                  float* __restrict__ accP,
                                  float* __restrict__ accN,
                                  int B_, int NU_) {
  const int wid = __builtin_amdgcn_readfirstlane(
      (int)(((long)blockIdx.x * blockDim.x + threadIdx.x) >> 5));
  const int lane = threadIdx.x & 31;
  if (wid >= 3 * B_) return;
  const int t = wid / B_;
  const int b = wid - t * B_;

  long row;
  float* acc;
  if (t == 0)      { row = userId[b];        acc = accU; }
  else if (t == 1) { row = NU_ + itemId[b];  acc = accP; }
  else             { row = NU_ + negId[b];   acc = accN; }

  const float4* s4 = (const float4*)(layer + row * D_EMB);
  float4* a4 = (float4*)(acc + (long)b * D_EMB);
  float4 s0 = s4[lane], s1 = s4[lane + 32];
  float4 d0 = a4[lane], d1 = a4[lane + 32];
  d0.x += s0.x; d0.y += s0.y; d0.z += s0.z; d0.w += s0.w;
  d1.x += s1.x; d1.y += s1.y; d1.z += s1.z; d1.w += s1.w;
  a4[lane] = d0;  a4[lane + 32] = d1;
}

// ----------------------------------------------- WMMA batched dot products
// One wave per 16 batch rows per (pos|neg) pair. C += A(16x4) x B(4x16) with
// A rows = acc_u tile, B cols = acc_{p,n} tile, looped over K = 256.
// Diagonal of the 16x16 f32 C tile = the 16 dot products.
// Layouts per ISA 7.12.2 (32-bit, wave32):
//   A 16x4 : lane L holds (M = L&15, K = (L>>4)*2 + v) in VGPR v  -> v2f
//   B 4x16 : lane L holds (N = L&15, K = (L>>4)*2 + v) in VGPR v  -> v2f
//   C/D    : lane L holds D[r + 8*(L>>4)][L&15] in VGPR r
// Diagonal: lanes 0..7 -> c[lane]; lanes 24..31 -> c[lane-24].
__global__ void wmma_scores_kernel(const float* __restrict__ accU,
                                   const float* __restrict__ accP,
                                   const float* __restrict__ accN,
                                   float* __restrict__ out, int B_) {
  const int wid = __builtin_amdgcn_readfirstlane(
      (int)(((long)blockIdx.x * blockDim.x + threadIdx.x) >> 5));
  const int lane = threadIdx.x & 31;
  const int tiles = B_ >> 4;                       // 256
  if (wid >= 2 * tiles) return;                    // wave-uniform: EXEC stays full
  const int pair = wid / tiles;                    // 0 = pos, 1 = neg
  const int tile = wid - pair * tiles;

  const float* U = accU + (long)tile * 16 * D_EMB;
  const float* P = (pair == 0 ? accP : accN) + (long)tile * 16 * D_EMB;

  const int mn = lane & 15;
  const int kp = (lane >> 4) * 2;

  v8f c = {0.f, 0.f, 0.f, 0.f, 0.f, 0.f, 0.f, 0.f};
  #pragma unroll 4
  for (int kb = 0; kb < D_EMB; kb += 4) {
    const float* ua = U + (long)mn * D_EMB + kb + kp;
    const float* pb = P + (long)mn * D_EMB + kb + kp;
    v2f a; a.x = ua[0]; a.y = ua[1];
    v2f b; b.x = pb[0]; b.y = pb[1];
    c = __builtin_amdgcn_wmma_f32_16x16x4_f32(
        /*neg_a=*/false, a, /*neg_b=*/false, b,
        /*c_mod=*/(short)0, c, /*reuse_a=*/false, /*reuse_b=*/false);
  }

  const int j = lane & 7;
  float v = (j == 0) ? c[0] : (j == 1) ? c[1] : (j == 2) ? c[2] : (j == 3) ? c[3]
          : (j == 4) ? c[4] : (j == 5) ? c[5] : (j == 6) ? c[6] : c[7];
  v *= 0.0625f;                                    // (acc/4) . (acc/4)

  float* sc = out + (long)pair * B_;
  if (lane < 8)        sc[tile * 16 + lane]        = v;   // diag rows 0..7
  else if (lane >= 24) sc[tile * 16 + (lane - 16)] = v;   // diag rows 8..15
}

// ---------------------------------------------------------------------------
extern "C" void kernel_launch(void* const* d_in, const int* in_sizes, int n_in,
                              void* d_out, int out_size, void* d_ws, size_t ws_size,
                              hipStream_t stream) {
  (void)n_in; (void)out_size; (void)ws_size;
  const float* uemb   = (const float*)d_in[0];
  const float* iemb   = (const float*)d_in[1];
  const float* vals   = (const float*)d_in[2];
  const int*   rows   = (const int*)d_in[3];
  const int*   cols   = (const int*)d_in[4];
  const int*   userId = (const int*)d_in[5];
  const int*   itemId = (const int*)d_in[6];
  const int*   negId  = (const int*)d_in[7];

  const int  NU  = in_sizes[0] / D_EMB;
  const int  NI  = in_sizes[1] / D_EMB;
  const long N   = (long)NU + NI;
  const int  nnz = in_sizes[3];
  const int  B   = in_sizes[5];

  float* out  = (float*)d_out;
  float* buf0 = (float*)d_ws;                       // N x D
  float* buf1 = buf0 + N * D_EMB;                   // N x D
  float* accU = buf1 + N * D_EMB;                   // B x D
  float* accP = accU + (long)B * D_EMB;             // B x D
  float* accN = accP + (long)B * D_EMB;             // B x D

  // buf0 = concat(user_emb, item_emb)
  hipMemcpyAsync(buf0, uemb, (size_t)in_sizes[0] * sizeof(float),
                 hipMemcpyDeviceToDevice, stream);
  hipMemcpyAsync(buf0 + (long)NU * D_EMB, iemb, (size_t)in_sizes[1] * sizeof(float),
                 hipMemcpyDeviceToDevice, stream);

  // acc = layer-0 (ego) rows; also emit ego outputs at d_out + 2B
  {
    const int waves  = 3 * B;
    const int blocks = (waves * 32 + 255) / 256;
    init_acc_ego_kernel<<<blocks, 256, 0, stream>>>(
        uemb, iemb, userId, itemId, negId, accU, accP, accN,
        out + 2l * B, B);
  }

  float* cur = buf0;
  float* nxt = buf1;
  const long n4 = N * D_EMB / 4;
  for (int layer = 0; layer < 3; ++layer) {
    zero_f4_kernel<<<2048, 256, 0, stream>>>((float4*)nxt, n4);

    const long waves  = ((long)nnz + CHUNK - 1) / CHUNK;
    const int  blocks = (int)((waves + 7) / 8);
    spmm_seg_kernel<<<blocks, 256, 0, stream>>>(rows, cols, vals, cur, nxt, nnz);

    const int gblocks = (3 * B * 32 + 255) / 256;
    gather_add_kernel<<<gblocks, 256, 0, stream>>>(
        nxt, userId, itemId, negId, accU, accP, accN, B, NU);

    float* t = cur; cur = nxt; nxt = t;
  }

  // scores = diag WMMA dots, scaled by 1/16
  {
    const int waves  = 2 * (B >> 4);
    const int blocks = (waves * 32 + 255) / 256;
    wmma_scores_kernel<<<blocks, 256, 0, stream>>>(accU, accP, accN, out, B);
  }
}